// SingleHeadAttention_15736760172860
// MI455X (gfx1250) — compile-verified
//
#include <hip/hip_runtime.h>

#define EMBED 768
#define ADIM  64
#define SEQ   4096
#define BATCH 4

typedef __attribute__((ext_vector_type(2))) float v2f;
typedef __attribute__((ext_vector_type(8))) float v8f;
typedef __attribute__((ext_vector_type(4))) unsigned int v4u;
typedef __attribute__((ext_vector_type(8))) int v8i;
typedef __attribute__((ext_vector_type(4))) int v4i;

// D(16x16,f32) = A(16x4,f32) x B(4x16,f32) + C
__device__ __forceinline__ v8f wmma_f32(v2f a, v2f b, v8f c) {
  return __builtin_amdgcn_wmma_f32_16x16x4_f32(
      /*neg_a=*/false, a, /*neg_b=*/false, b,
      /*c_mod=*/(short)0, c, /*reuse_a=*/false, /*reuse_b=*/false);
}

// ---------------------------------------------------------------------------
// TDM: async DMA of a 16x64 f32 tile (row-major, row stride 64 elements in
// memory) into LDS with a 4-dword pad after every 64 dwords -> LDS row
// stride 68 floats (bank-conflict-free B-operand reads).
// D# per CDNA5 ISA (group0 128b, group1 256b, 2D).
// 6-arg builtin form (clang-23 / therock-10.0 headers).
// ---------------------------------------------------------------------------
__device__ __forceinline__ void tdm_load_tile16x64(unsigned int lds_addr,
                                                   const float* gsrc) {
  unsigned long long ga = (unsigned long long)(uintptr_t)gsrc;
  v4u g0;
  g0.x = 1u;                                    // count=1, user mode
  g0.y = lds_addr;                              // lds_addr [63:32]
  g0.z = (unsigned int)ga;                      // global_addr [95:64]
  g0.w = ((unsigned int)(ga >> 32) & 0x01FFFFFFu) | 0x80000000u; // addr hi + type=2

  v8i g1;
  g1[0] = (int)((2u << 16)     // data_size = 4 bytes
              | (1u << 20)     // pad_enable
              | (5u << 22)     // pad_interval code 5 -> every 64 dwords
              | (3u << 25));   // pad_amount  code 3 -> 4 dwords
  g1[1] = (int)(64u << 16);    // ab_addr=0 | tensor_dim0.lo16 = 64
  g1[2] = (int)(16u << 16);    // tensor_dim0.hi16=0 | tensor_dim1.lo16 = 16
  g1[3] = (int)(64u << 16);    // tensor_dim1.hi16=0 | tile_dim0 = 64
  g1[4] = 16;                  // tile_dim1 = 16 | tile_dim2 = 0
  g1[5] = 64;                  // tensor_dim0_stride.lo32 = 64 elements
  g1[6] = 0;                   // stride0.hi16 | tensor_dim1_stride.lo16
  g1[7] = 0;

  v4i z4 = {0, 0, 0, 0};       // groups 2/3: unused (2D tensor)
  v8i z8 = {0, 0, 0, 0, 0, 0, 0, 0};
  __builtin_amdgcn_tensor_load_to_lds(g0, g1, z4, z4, z8, 0);
}

// ---------------------------------------------------------------------------
// Stage 1: Q/K/V = X @ W^T  (X: [B,S,768], W: [64,768], Y: [B,S,64])
// One wave computes one 16x64 output tile of one of {Q,K,V}.
// ---------------------------------------------------------------------------
__global__ __launch_bounds__(256) void qkv_kernel(
    const float* __restrict__ X,
    const float* __restrict__ Wq, const float* __restrict__ Wk,
    const float* __restrict__ Wv,
    float* __restrict__ Q, float* __restrict__ K, float* __restrict__ V) {
  const int lane = threadIdx.x & 31;
  const int wave = threadIdx.x >> 5;
  const int gw   = blockIdx.x * 8 + wave;   // 0..3071
  const int mat  = gw >> 10;                // 1024 tiles per matrix
  const int tile = gw & 1023;
  const int b    = tile >> 8;               // 256 tiles per batch
  const int row0 = (tile & 255) << 4;

  const float* W = (mat == 0) ? Wq : (mat == 1) ? Wk : Wv;
  float*       Y = (mat == 0) ? Q  : (mat == 1) ? K  : V;

  const int n = lane & 15;
  const int h = lane >> 4;

  const float* xrow = X + ((size_t)b * SEQ + row0 + n) * EMBED + 2 * h;

  v8f acc[4];
  acc[0] = {}; acc[1] = {}; acc[2] = {}; acc[3] = {};

  for (int c = 0; c < EMBED / 4; ++c) {
    v2f a = *(const v2f*)(xrow + 4 * c);
#pragma unroll
    for (int t = 0; t < 4; ++t) {
      v2f bb = *(const v2f*)(W + (size_t)(t * 16 + n) * EMBED + 4 * c + 2 * h);
      acc[t] = wmma_f32(a, bb, acc[t]);
    }
  }

  float* yb = Y + (size_t)b * SEQ * ADIM;
#pragma unroll
  for (int t = 0; t < 4; ++t)
#pragma unroll
    for (int r = 0; r < 8; ++r)
      yb[(size_t)(row0 + r + 8 * h) * ADIM + t * 16 + n] = acc[t][r];
}

// ---------------------------------------------------------------------------
// Stage 2: causal flash attention. Workgroup = 8 waves = 128 query rows.
// K/V blocks double-buffered in LDS via TDM; wave 0 drives the DMA.
// ---------------------------------------------------------------------------
__global__ __launch_bounds__(256) void fa_kernel(
    const float* __restrict__ Q, const float* __restrict__ K,
    const float* __restrict__ V, float* __restrict__ O) {
  __shared__ __align__(16) float Ksh[2][16 * 68];   // TDM-padded stride 68
  __shared__ __align__(16) float Vsh[2][16 * 68];
  __shared__ __align__(16) float Psh[8][16 * 18];   // per-wave P bounce

  const int lane  = threadIdx.x & 31;
  const int wave  = threadIdx.x >> 5;
  const int b     = blockIdx.y;
  const int qb0   = blockIdx.x * 128;
  const int qbase = qb0 + wave * 16;
  const int n = lane & 15;
  const int h = lane >> 4;

  // Preload Q tile in A-layout chunks (32 VGPRs)
  const float* qrow = Q + ((size_t)b * SEQ + qbase + n) * ADIM + 2 * h;
  v2f qa[16];
#pragma unroll
  for (int c = 0; c < 16; ++c) qa[c] = *(const v2f*)(qrow + 4 * c);

  v8f acc[4];
  acc[0] = {}; acc[1] = {}; acc[2] = {}; acc[3] = {};
  float mrun[8], lrun[8];
#pragma unroll
  for (int r = 0; r < 8; ++r) { mrun[r] = -3.0e30f; lrun[r] = 0.0f; }

  const int   nblocks = (qb0 + 128) / 16;
  const float scale   = 0.125f;  // 1/sqrt(64)

  const float* kb0 = K + (size_t)b * SEQ * ADIM;
  const float* vb0 = V + (size_t)b * SEQ * ADIM;

  // Prologue: DMA block 0 into buffer 0 (wave 0 drives TDM; TENSORcnt is
  // per-wave, TDM ignores EXEC).
  if (wave == 0) {
    tdm_load_tile16x64((unsigned int)(uintptr_t)&Ksh[0][0], kb0);
    tdm_load_tile16x64((unsigned int)(uintptr_t)&Vsh[0][0], vb0);
    __builtin_amdgcn_s_wait_tensorcnt(0);
  }
  __syncthreads();

  for (int j = 0; j < nblocks; ++j) {
    const int kbase = j * 16;
    const int cur   = j & 1;

    // Issue DMA for the next block into the other buffer (overlaps compute).
    if (wave == 0 && (j + 1) < nblocks) {
      const int nxt = (j + 1) & 1;
      tdm_load_tile16x64((unsigned int)(uintptr_t)&Ksh[nxt][0],
                         kb0 + (size_t)(kbase + 16) * ADIM);
      tdm_load_tile16x64((unsigned int)(uintptr_t)&Vsh[nxt][0],
                         vb0 + (size_t)(kbase + 16) * ADIM);
    }

    if (kbase <= qbase) {   // wave-uniform: EXEC stays all-ones for WMMA
      // ---- scores S = Q * K^T  (K-dim 64 -> 16 wmma) ----
      v8f s = {};
#pragma unroll
      for (int c = 0; c < 16; ++c) {
        v2f bb = *(const v2f*)(&Ksh[cur][n * 68 + 4 * c + 2 * h]);
        s = wmma_f32(qa[c], bb, s);
      }
      // ---- scale + causal mask (only diagonal block is partial) ----
      const bool diag = (kbase == qbase);
#pragma unroll
      for (int r = 0; r < 8; ++r) {
        const int m = r + 8 * h;
        float sv = s[r] * scale;
        if (diag && (n > m)) sv = -3.0e30f;
        s[r] = sv;
      }
      // ---- online softmax (row = r + 8h, 16 cols across lane-half) ----
#pragma unroll
      for (int r = 0; r < 8; ++r) {
        float rm = s[r];
        rm = fmaxf(rm, __shfl_xor(rm, 8, 32));
        rm = fmaxf(rm, __shfl_xor(rm, 4, 32));
        rm = fmaxf(rm, __shfl_xor(rm, 2, 32));
        rm = fmaxf(rm, __shfl_xor(rm, 1, 32));
        const float mnew = fmaxf(mrun[r], rm);
        const float p    = __expf(s[r] - mnew);
        float rs = p;
        rs += __shfl_xor(rs, 8, 32);
        rs += __shfl_xor(rs, 4, 32);
        rs += __shfl_xor(rs, 2, 32);
        rs += __shfl_xor(rs, 1, 32);
        const float alpha = __expf(mrun[r] - mnew);
        lrun[r] = lrun[r] * alpha + rs;
        mrun[r] = mnew;
#pragma unroll
        for (int t = 0; t < 4; ++t) acc[t][r] *= alpha;
        Psh[wave][(r + 8 * h) * 18 + n] = p;  // re-layout P for A-operand
      }
      asm volatile("s_wait_dscnt 0" ::: "memory");
      // ---- O += P(16x16) * V(16x64) ----
#pragma unroll
      for (int t = 0; t < 4; ++t) {
#pragma unroll
        for (int c = 0; c < 4; ++c) {
          v2f pa = *(const v2f*)(&Psh[wave][n * 18 + 4 * c + 2 * h]);
          v2f bb;
          bb.x = Vsh[cur][(4 * c + 2 * h) * 68 + t * 16 + n];
          bb.y = Vsh[cur][(4 * c + 1 + 2 * h) * 68 + t * 16 + n];
          acc[t] = wmma_f32(pa, bb, acc[t]);
        }
      }
    }

    // Next block's DMA must be complete (and this block's LDS free) before
    // anyone proceeds.
    if (wave == 0) __builtin_amdgcn_s_wait_tensorcnt(0);
    __syncthreads();
  }

  // ---- epilogue: normalize and store ----
  float* ob = O + (size_t)b * SEQ * ADIM;
#pragma unroll
  for (int t = 0; t < 4; ++t)
#pragma unroll
    for (int r = 0; r < 8; ++r)
      ob[(size_t)(qbase + r + 8 * h) * ADIM + t * 16 + n] =
          acc[t][r] / lrun[r];
}

extern "C" void kernel_launch(void* const* d_in, const int* in_sizes, int n_in,
                              void* d_out, int out_size, void* d_ws,
                              size_t ws_size, hipStream_t stream) {
  const float* X  = (const float*)d_in[0];
  const float* Wq = (const float*)d_in[1];
  const float* Wk = (const float*)d_in[2];
  const float* Wv = (const float*)d_in[3];

  const size_t qkv = (size_t)BATCH * SEQ * ADIM;  // 1M floats = 4MB each
  float* Q = (float*)d_ws;
  float* K = Q + qkv;
  float* V = K + qkv;

  qkv_kernel<<<dim3(384), 256, 0, stream>>>(X, Wq, Wk, Wv, Q, K, V);
  fa_kernel<<<dim3(SEQ / 128, BATCH), 256, 0, stream>>>(Q, K, V,
                                                        (float*)d_out);
}